// Mapping_and_Shared_Centroids_55508157333549
// MI455X (gfx1250) — compile-verified
//
#include <hip/hip_runtime.h>
#include <hip/hip_bf16.h>

typedef __attribute__((ext_vector_type(2))) float v2f;
typedef __attribute__((ext_vector_type(8))) float v8f;

#define N_TOK   32768
#define TOKDIM  1024
#define HIDDIM  1024
#define SENSED  256
#define KCENT   512

// ---------------------------------------------------------------------------
// GEMM  C[M,N] = act(A[M,K] @ B[K,N]),  f32 WMMA 16x16x4.  N,K compile-time
// so all global accesses are one base address + immediate offsets.
// Block: 128 threads (4 waves). Block tile 128x64; wave w owns rows
// [32w,32w+32) x 64 cols = 8 v8f accumulators. K staged through double-
// buffered LDS in steps of 32, with staging split into two phases
// interleaved with the k-loop halves (buffer cur^1 is write-safe during
// compute of cur), so at most 32 staged VGPRs are live at once and only
// one barrier per stage is needed. A rows padded to 36 floats (16B-aligned,
// conflict-free b64 frag reads). B packed as K-pair float2 rows padded to
// 80 so each fragment is ONE aligned ds_load_b64 and the two half-waves
// hit disjoint bank halves. M%128==0, N%64==0, K%32==0.
// ---------------------------------------------------------------------------
template <bool TANH, int N, int K>
__global__ __launch_bounds__(128, 1)
void gemm_wmma_f32(const float* __restrict__ A, const float* __restrict__ B,
                   float* __restrict__ C) {
    constexpr int AP = 36;            // A LDS row stride (floats)
    constexpr int BP = 80;            // B LDS row stride (float2), 80%32==16
    __shared__ float As[2][128 * AP];
    __shared__ v2f  BsP[2][16 * BP];

    const int tid  = threadIdx.x;
    const int wave = tid >> 5;
    const int lane = tid & 31;
    const int l15  = lane & 15;
    const int h    = (lane >> 4) * 2;           // K sub-select (0 or 2)
    const int bm   = blockIdx.y * 128;
    const int bn   = blockIdx.x * 64;

    // single per-thread base pointers; per-i deltas are compile-time consts
    const float* aptr = A + (size_t)(bm + (tid >> 3)) * K + 4 * (tid & 7);
    const float* bptr = B + (size_t)(2 * (tid >> 5)) * N + bn + 2 * (tid & 31);

    float4 va[4];                                // half of the A tile at a time
    v2f    g0[4], g1[4];

    auto loadA = [&](int half) {
        #pragma unroll
        for (int i = 0; i < 4; ++i)
            va[i] = *(const float4*)(aptr + (size_t)(16 * (4 * half + i)) * K);
    };
    auto storeA = [&](int buf, int half) {
        #pragma unroll
        for (int i = 0; i < 4; ++i) {
            int item = tid + 128 * (4 * half + i);
            *(float4*)&As[buf][(item >> 3) * AP + 4 * (item & 7)] = va[i];
        }
    };
    auto loadB = [&]() {
        #pragma unroll
        for (int i = 0; i < 4; ++i) {
            g0[i] = *(const v2f*)(bptr + (size_t)(8 * i) * N);      // k=2r
            g1[i] = *(const v2f*)(bptr + (size_t)(8 * i) * N + N);  // k=2r+1
        }
    };
    auto storeB = [&](int buf) {
        #pragma unroll
        for (int i = 0; i < 4; ++i) {
            int item = tid + 128 * i;
            int r = item >> 5, c2 = item & 31;
            v2f lo, hi;
            lo.x = g0[i].x; lo.y = g1[i].x;      // col 2c2:   (B[k], B[k+1])
            hi.x = g0[i].y; hi.y = g1[i].y;      // col 2c2+1
            BsP[buf][r * BP + 2 * c2]     = lo;
            BsP[buf][r * BP + 2 * c2 + 1] = hi;
        }
    };
    auto advance = [&]() { aptr += 32; bptr += (size_t)32 * N; };

    const int mA0 = (32 * wave + l15) * AP;      // A frag row offsets (floats)
    const int mA1 = mA0 + 16 * AP;

    v8f acc[2][4] = {};
    auto compute16 = [&](const float* as, const v2f* bs, int kkbase) {
        #pragma unroll
        for (int kk2 = 0; kk2 < 16; kk2 += 4) {
            const int kk = kkbase + kk2;
            v2f a0 = *(const v2f*)&as[mA0 + kk + h];
            v2f a1 = *(const v2f*)&as[mA1 + kk + h];
            const int kp = (kk >> 1) + (lane >> 4);
            #pragma unroll
            for (int j = 0; j < 4; ++j) {
                v2f b = bs[kp * BP + 16 * j + l15];
                acc[0][j] = __builtin_amdgcn_wmma_f32_16x16x4_f32(
                    false, a0, false, b, (short)0, acc[0][j], false, false);
                acc[1][j] = __builtin_amdgcn_wmma_f32_16x16x4_f32(
                    false, a1, false, b, (short)0, acc[1][j], false, false);
            }
        }
    };

    // prologue: fill buffer 0 (low pressure: halves staged sequentially)
    loadB();  loadA(0);
    storeB(0); storeA(0, 0);
    loadA(1);  storeA(0, 1);
    advance();
    __syncthreads();

    constexpr int S = K / 32;
    #pragma unroll 1
    for (int s = 0; s < S; ++s) {
        const int cur = s & 1;
        const float* as = As[cur];
        const v2f*   bs = BsP[cur];
        const bool more = (s + 1 < S);

        if (more) { loadB(); loadA(0); }         // loads fly over compute
        compute16(as, bs, 0);                    // kk = 0..12
        if (more) {                              // cur^1 is write-safe here
            storeB(cur ^ 1); storeA(cur ^ 1, 0);
            loadA(1);
        }
        compute16(as, bs, 16);                   // kk = 16..28
        if (more) { storeA(cur ^ 1, 1); advance(); }
        __syncthreads();                         // single barrier per stage
    }

    // C/D layout: vgpr r, lanes0-15 -> M=r, lanes16-31 -> M=r+8
    const int mbase0 = bm + 32 * wave + ((lane >> 4) ? 8 : 0);
    float* cptr = C + (size_t)mbase0 * N + bn + l15;
    #pragma unroll
    for (int mi = 0; mi < 2; ++mi) {
        #pragma unroll
        for (int j = 0; j < 4; ++j) {
            #pragma unroll
            for (int r = 0; r < 8; ++r) {
                float v = acc[mi][j][r];
                if (TANH) v = tanhf(v);
                cptr[(size_t)(16 * mi + r) * N + 16 * j] = v;
            }
        }
    }
}

// ---------------------------------------------------------------------------
// centroids = tanh(raw); also emit transposed copy for the score GEMM
// ---------------------------------------------------------------------------
__global__ void centroid_prep(const float* __restrict__ raw,
                              float* __restrict__ cent_out,
                              float* __restrict__ cT) {
    int i = blockIdx.x * blockDim.x + threadIdx.x;      // 512*256 total
    if (i < KCENT * SENSED) {
        float c = tanhf(raw[i]);
        cent_out[i] = c;
        int k = i >> 8, d = i & 255;
        cT[(size_t)d * KCENT + k] = c;                  // [256,512]
    }
}

// deterministic per-centroid squared norms (no float atomics)
__global__ void centroid_norms(const float* __restrict__ cent,
                               float* __restrict__ c2) {
    int k = blockIdx.x * blockDim.x + threadIdx.x;
    if (k < KCENT) {
        float s = 0.f;
        const float* row = cent + (size_t)k * SENSED;
        for (int d = 0; d < SENSED; ++d) s += row[d] * row[d];
        c2[k] = s;
    }
}

__global__ void zero_ints(int* __restrict__ p, int n) {
    int i = blockIdx.x * blockDim.x + threadIdx.x;
    if (i < n) p[i] = 0;
}

// ---------------------------------------------------------------------------
// argmin_k ( c2[k] - 2*score[n,k] )  (e2 is constant per token -> dropped).
// One wave per token; smaller-k wins ties => first-occurrence argmin.
// ---------------------------------------------------------------------------
__global__ void assign_kernel(const float* __restrict__ scores,
                              const float* __restrict__ c2,
                              int* __restrict__ sc, int* __restrict__ counts,
                              float* __restrict__ out_sense,
                              float* __restrict__ out_word, float wordval) {
    int token = blockIdx.x * (blockDim.x >> 5) + (threadIdx.x >> 5);
    int lane  = threadIdx.x & 31;
    if (token >= N_TOK) return;
    const float* row = scores + (size_t)token * KCENT;

    float best = 3.4e38f;
    int bestk  = 0x7fffffff;
    for (int k = lane; k < KCENT; k += 32) {
        float v = c2[k] - 2.0f * row[k];
        if (v < best) { best = v; bestk = k; }           // k increasing per lane
    }
    #pragma unroll
    for (int off = 16; off > 0; off >>= 1) {
        float ov = __shfl_xor(best, off, 32);
        int   ok = __shfl_xor(bestk, off, 32);
        if (ov < best || (ov == best && ok < bestk)) { best = ov; bestk = ok; }
    }
    if (lane == 0) {
        sc[token] = bestk;
        atomicAdd(&counts[bestk], 1);                    // int atomic: deterministic
        out_sense[token] = (float)bestk;
        out_word[token]  = wordval;
    }
}

// ---------------------------------------------------------------------------
// Stable counting-sort permutation == jnp.argsort(sc, stable=True).
// ---------------------------------------------------------------------------
__global__ __launch_bounds__(512)
void order_kernel(const int* __restrict__ sc, const int* __restrict__ counts,
                  float* __restrict__ out_order, float* __restrict__ out_counts) {
    __shared__ int offs[KCENT];
    int k = threadIdx.x;
    int c = counts[k];
    out_counts[k] = (float)c;
    offs[k] = c;
    __syncthreads();
    for (int d = 1; d < KCENT; d <<= 1) {               // Hillis-Steele inclusive
        int v = (k >= d) ? offs[k - d] : 0;
        __syncthreads();
        offs[k] += v;
        __syncthreads();
    }
    int off = offs[k] - c;                               // exclusive prefix
    for (int n = 0; n < N_TOK; ++n)
        if (sc[n] == k) out_order[off++] = (float)n;
}

// ---------------------------------------------------------------------------
extern "C" void kernel_launch(void* const* d_in, const int* in_sizes, int n_in,
                              void* d_out, int out_size, void* d_ws, size_t ws_size,
                              hipStream_t stream) {
    const float* text_emb = (const float*)d_in[0];
    const float* image_emb= (const float*)d_in[1];
    const float* text_W1  = (const float*)d_in[2];
    const float* text_W2  = (const float*)d_in[3];
    const float* text_W3  = (const float*)d_in[4];
    const float* image_W1 = (const float*)d_in[5];
    const float* image_W2 = (const float*)d_in[6];
    const float* image_W3 = (const float*)d_in[7];
    const float* cen_raw  = (const float*)d_in[8];

    // ---- output layout (flat f32, reference return order) ----
    float* out        = (float*)d_out;
    float* cent_out   = out;                                   // [512,256]
    float* se_out     = cent_out + KCENT * SENSED;             // [2N,256]
    float* word_out   = se_out + (size_t)2 * N_TOK * SENSED;   // [2N]
    float* sense_out  = word_out + 2 * N_TOK;                  // [2N]
    float* order_t_o  = sense_out + 2 * N_TOK;                 // [N]
    float* counts_t_o = order_t_o + N_TOK;                     // [512]
    float* order_i_o  = counts_t_o + KCENT;                    // [N]
    float* counts_i_o = order_i_o + N_TOK;                     // [512]

    // ---- workspace layout ----
    float* wsf     = (float*)d_ws;
    float* cT      = wsf;                       // [256,512]
    float* c2      = wsf + 131072;              // [512]
    int*   cnt_t   = (int*)(wsf + 131584);      // [512]
    int*   cnt_i   = cnt_t + KCENT;             // [512]
    int*   sc_t    = cnt_i + KCENT;             // [N]
    int*   sc_i    = sc_t + N_TOK;              // [N]
    float* h1      = wsf + 198144;              // [N,1024] (aliased by scores)
    float* h2      = h1 + (size_t)N_TOK * HIDDIM;

    centroid_prep <<<(KCENT * SENSED + 255) / 256, 256, 0, stream>>>(cen_raw, cent_out, cT);
    centroid_norms<<<2, 256, 0, stream>>>(cent_out, c2);
    zero_ints     <<<1, 1024, 0, stream>>>(cnt_t, 2 * KCENT);

    const dim3 gHid (HIDDIM / 64, N_TOK / 128);   // (16, 256)
    const dim3 gSen (SENSED / 64, N_TOK / 128);   // ( 4, 256)
    const dim3 gScr (KCENT  / 64, N_TOK / 128);   // ( 8, 256)

    for (int modal = 0; modal < 2; ++modal) {
        const float* emb = modal ? image_emb : text_emb;
        const float* W1  = modal ? image_W1  : text_W1;
        const float* W2  = modal ? image_W2  : text_W2;
        const float* W3  = modal ? image_W3  : text_W3;
        float* se   = se_out + (size_t)modal * N_TOK * SENSED;
        int*   sc   = modal ? sc_i  : sc_t;
        int*   cnt  = modal ? cnt_i : cnt_t;
        float* ordo = modal ? order_i_o  : order_t_o;
        float* cnto = modal ? counts_i_o : counts_t_o;

        gemm_wmma_f32<true,  HIDDIM, TOKDIM><<<gHid, 128, 0, stream>>>(emb, W1, h1);
        gemm_wmma_f32<true,  HIDDIM, HIDDIM><<<gHid, 128, 0, stream>>>(h1,  W2, h2);
        gemm_wmma_f32<true,  SENSED, HIDDIM><<<gSen, 128, 0, stream>>>(h2,  W3, se);
        // scores[n,k] = se_n . c_k   (reuses h1)
        gemm_wmma_f32<false, KCENT,  SENSED><<<gScr, 128, 0, stream>>>(se,  cT, h1);

        assign_kernel<<<N_TOK / 8, 256, 0, stream>>>(h1, c2, sc, cnt,
                                                     sense_out + (size_t)modal * N_TOK,
                                                     word_out  + (size_t)modal * N_TOK,
                                                     (float)modal);
        order_kernel<<<1, 512, 0, stream>>>(sc, cnt, ordo, cnto);
    }
}